// A3C_Model_27848567947758
// MI455X (gfx1250) — compile-verified
//
#include <hip/hip_runtime.h>

typedef __attribute__((ext_vector_type(2))) float v2f;
typedef __attribute__((ext_vector_type(8))) float v8f;

#define F_IN   65536
#define NND    100
#define MP     112      // padded node count (7 x 16)
#define COUT   60
#define NPAD   64       // padded out channels (4 x 16)
#define NEDGE  2000
#define KS     128      // K per chunk
#define LDA    132      // LDS A-tile row stride (pad: 132 % 64 != 0)
#define NMAT   6        // {actor,critic} x {W0,W1,W2}
#define NBLK   128      // split-K blocks
#define CHUNKS (F_IN / KS)        // 512
#define YSZ    (NMAT * MP * NPAD) // 43008

// ws layout (float offsets)
#define OFF_L    0
#define OFF_T2   (MP * MP)            // 12544
#define OFF_Y    (2 * MP * MP)        // 25088
#define OFF_DEG  (OFF_Y + YSZ)        // 68096
#define OFF_PART 68608                // NBLK * YSZ floats follow (~21 MB)

// ---------------------------------------------------------------------------
// Kernel 1: build L_hat (dense 112x112, zero padded) and T2 = 2*L^2 - I
// ---------------------------------------------------------------------------
__global__ void prep_kernel(const int* __restrict__ eidx, float* __restrict__ ws) {
    float* L   = ws + OFF_L;
    float* T2  = ws + OFF_T2;
    float* deg = ws + OFF_DEG;
    __shared__ float dis[MP];
    __shared__ float Ls[MP * MP];   // ~50 KB
    const int tid = threadIdx.x;

    for (int i = tid; i < MP * MP; i += 256) L[i] = 0.0f;
    for (int i = tid; i < MP;      i += 256) deg[i] = 0.0f;
    __syncthreads();
    __threadfence();

    for (int e = tid; e < NEDGE; e += 256) {
        int s = eidx[e];
        atomicAdd(&deg[s], 1.0f);
    }
    __syncthreads();
    __threadfence();

    for (int i = tid; i < MP; i += 256) {
        float d = deg[i];
        dis[i] = (d > 0.0f) ? rsqrtf(d) : 0.0f;
    }
    __syncthreads();

    for (int e = tid; e < NEDGE; e += 256) {
        int s = eidx[e];
        int d = eidx[NEDGE + e];
        atomicAdd(&L[d * MP + s], -dis[s] * dis[d]);
    }
    __syncthreads();
    __threadfence();

    for (int i = tid; i < MP * MP; i += 256) Ls[i] = L[i];
    __syncthreads();

    for (int idx = tid; idx < MP * MP; idx += 256) {
        int i = idx / MP, j = idx % MP;
        float s = 0.0f;
        for (int t = 0; t < MP; ++t) s += Ls[i * MP + t] * Ls[t * MP + j];
        T2[idx] = 2.0f * s - ((i == j) ? 1.0f : 0.0f);
    }
}

// ---------------------------------------------------------------------------
// Kernel 2: Yk_partial = x @ Wk for 6 weight matrices, split-K with WMMA f32.
// Block: 768 threads = 24 waves; wave w -> (mat = w>>2, ntile = w&3).
// Each wave accumulates 7 m-tiles (full padded M=112) in v8f registers.
// B columns >= 60 read a clamped duplicate of column 59 (never consumed),
// keeping the inner loop free of exec-mask divergence.
// ---------------------------------------------------------------------------
__global__ void __launch_bounds__(768) gemm_kernel(const float* __restrict__ x,
                                                   const float* __restrict__ aw,
                                                   const float* __restrict__ cw,
                                                   float* __restrict__ part) {
    __shared__ float As[MP * LDA];  // 57.8 KB

    const int tid   = threadIdx.x;
    const int lane  = tid & 31;
    const int wid   = tid >> 5;     // 0..23
    const int mat   = wid >> 2;     // 0..5
    const int ntile = wid & 3;      // 0..3
    const int lhi   = lane >> 4;    // 0: K{0,1}, 1: K{2,3}
    const int nloc  = lane & 15;
    const int n     = ntile * 16 + nloc;
    const int nc    = (n < COUT) ? n : (COUT - 1);   // clamp: no divergence, no OOB
    const float* Bp = (mat < 3) ? (aw + mat * (F_IN * COUT))
                                : (cw + (mat - 3) * (F_IN * COUT));

    v8f acc[7] = {};

    for (int ch = (int)blockIdx.x; ch < CHUNKS; ch += (int)gridDim.x) {
        const int kbase = ch * KS;
        __syncthreads();
        // Stage A tile [112 x 128] (rows >= 100 are zero), coalesced float4
        for (int idx = tid; idx < MP * (KS / 4); idx += 768) {
            int row = idx >> 5;   // 32 float4 per row
            int c4  = idx & 31;
            float4 v = make_float4(0.0f, 0.0f, 0.0f, 0.0f);
            if (row < NND)
                v = *(const float4*)(x + row * F_IN + kbase + c4 * 4);
            *((float4*)&As[row * LDA + c4 * 4]) = v;
        }
        __syncthreads();

        // Per-chunk base pointer: inner-loop B addresses become immediate
        // offsets (kk*COUT*4 <= 29.8 KB fits the 24-bit ioffset).
        const float* bq = Bp + (size_t)kbase * COUT + lhi * 2 * COUT + nc;

        for (int kk = 0; kk < KS; kk += 4) {
            v2f b;
            b.x = bq[kk * COUT];          // K = kk   (+2 for high lanes)
            b.y = bq[kk * COUT + COUT];   // K = kk+1 (+2 for high lanes)
#pragma unroll
            for (int mt = 0; mt < 7; ++mt) {
                const float* ap = &As[(mt * 16 + nloc) * LDA + kk + lhi * 2];
                v2f a;
                a.x = ap[0];
                a.y = ap[1];
                acc[mt] = __builtin_amdgcn_wmma_f32_16x16x4_f32(
                    false, a, false, b, (short)0, acc[mt], false, false);
            }
        }
    }

    // Write this block's partial [mat][112][64]
    float* dst = part + (size_t)blockIdx.x * YSZ + mat * (MP * NPAD);
#pragma unroll
    for (int mt = 0; mt < 7; ++mt) {
#pragma unroll
        for (int r = 0; r < 8; ++r) {
            int m = mt * 16 + r + lhi * 8;  // C/D layout: VGPR r -> M=r (+8 high lanes)
            dst[m * NPAD + n] = acc[mt][r];
        }
    }
}

// ---------------------------------------------------------------------------
// Kernel 3: reduce split-K partials -> Y[6][112][64]
// ---------------------------------------------------------------------------
__global__ void reduce_kernel(float* __restrict__ ws) {
    int i = (int)blockIdx.x * 256 + (int)threadIdx.x;
    if (i >= YSZ) return;
    const float* p = ws + OFF_PART + i;
    float s = 0.0f;
    for (int b = 0; b < NBLK; ++b) s += p[(size_t)b * YSZ];
    ws[OFF_Y + i] = s;
}

// ---------------------------------------------------------------------------
// Kernel 4: combine (out = Y0 + L@Y1 + T2@Y2 + b), tanh, build states, FCs
// ---------------------------------------------------------------------------
__global__ void tail_kernel(const float* __restrict__ ab, const float* __restrict__ cb,
                            const float* __restrict__ afcw, const float* __restrict__ afcb,
                            const float* __restrict__ cfcw, const float* __restrict__ cfcb,
                            const float* __restrict__ cpu, const float* __restrict__ bw,
                            const float* __restrict__ pend,
                            const float* __restrict__ ws, float* __restrict__ out) {
    __shared__ float st[2][6016];   // two 6003-element state vectors
    __shared__ float red[256];
    const float* L  = ws + OFF_L;
    const float* T2 = ws + OFF_T2;
    const float* Y  = ws + OFF_Y;
    const int tid = threadIdx.x;

    for (int head = 0; head < 2; ++head) {
        const float* bias = head ? cb : ab;
        const float* Y0 = Y + (head * 3 + 0) * (MP * NPAD);
        const float* Y1 = Y + (head * 3 + 1) * (MP * NPAD);
        const float* Y2 = Y + (head * 3 + 2) * (MP * NPAD);
        for (int idx = tid; idx < NND * COUT; idx += 256) {
            int m = idx / COUT, c = idx % COUT;
            float s = Y0[m * NPAD + c] + bias[c];
            for (int j = 0; j < MP; ++j)
                s += L[m * MP + j] * Y1[j * NPAD + c] +
                     T2[m * MP + j] * Y2[j * NPAD + c];
            st[head][idx] = tanhf(s);
        }
    }
    if (tid == 0) {
        st[0][6000] = cpu[0]; st[0][6001] = bw[0]; st[0][6002] = pend[0];
        st[1][6000] = cpu[0]; st[1][6001] = bw[0]; st[1][6002] = pend[0];
    }
    __syncthreads();

    // critic value: parallel partial sums + tree reduce
    float v = 0.0f;
    for (int i = tid; i < 6003; i += 256) v += st[1][i] * cfcw[i];
    red[tid] = v;
    __syncthreads();
    for (int off = 128; off > 0; off >>= 1) {
        if (tid < off) red[tid] += red[tid + off];
        __syncthreads();
    }
    if (tid == 0) out[100] = red[0] + cfcb[0];

    // actor logits: one column per thread (coalesced over j)
    for (int j = tid; j < 100; j += 256) {
        float s = afcb[j];
        for (int i = 0; i < 6003; ++i) s += st[0][i] * afcw[i * 100 + j];
        out[j] = s;
    }
}

// ---------------------------------------------------------------------------
extern "C" void kernel_launch(void* const* d_in, const int* in_sizes, int n_in,
                              void* d_out, int out_size, void* d_ws, size_t ws_size,
                              hipStream_t stream) {
    (void)in_sizes; (void)n_in; (void)out_size; (void)ws_size;
    const float* x    = (const float*)d_in[0];
    const int*   eidx = (const int*)d_in[1];
    const float* cpu  = (const float*)d_in[2];
    const float* bw   = (const float*)d_in[3];
    const float* pend = (const float*)d_in[4];
    const float* aw   = (const float*)d_in[5];
    const float* ab   = (const float*)d_in[6];
    const float* cw   = (const float*)d_in[7];
    const float* cbb  = (const float*)d_in[8];
    const float* afcw = (const float*)d_in[9];
    const float* afcb = (const float*)d_in[10];
    const float* cfcw = (const float*)d_in[11];
    const float* cfcb = (const float*)d_in[12];
    float* ws  = (float*)d_ws;
    float* out = (float*)d_out;

    prep_kernel<<<1, 256, 0, stream>>>(eidx, ws);
    gemm_kernel<<<NBLK, 768, 0, stream>>>(x, aw, cw, ws + OFF_PART);
    reduce_kernel<<<(YSZ + 255) / 256, 256, 0, stream>>>(ws);
    tail_kernel<<<1, 256, 0, stream>>>(ab, cbb, afcw, afcb, cfcw, cfcb,
                                       cpu, bw, pend, ws, out);
}